// GCN_88802743812231
// MI455X (gfx1250) — compile-verified
//
#include <hip/hip_runtime.h>
#include <hip/hip_bf16.h>

typedef float v2f __attribute__((ext_vector_type(2)));
typedef float v8f __attribute__((ext_vector_type(8)));

// ---------------------------------------------------------------- degree ----
__global__ void deg_init_kernel(float* __restrict__ deg, int N) {
    int i = blockIdx.x * blockDim.x + threadIdx.x;
    if (i < N) deg[i] = 1.0f;   // self loop
}

__global__ void deg_accum_kernel(const long long* __restrict__ eidx,
                                 float* __restrict__ deg, int E) {
    int e = blockIdx.x * blockDim.x + threadIdx.x;
    if (e < E) {
        int d = (int)eidx[(size_t)E + e];   // dst row of edge_index
        atomicAdd(&deg[d], 1.0f);
    }
}

__global__ void dinv_kernel(float* __restrict__ deg, int N) {
    int i = blockIdx.x * blockDim.x + threadIdx.x;
    if (i < N) deg[i] = rsqrtf(deg[i]);     // in-place: deg -> dinv
}

// ------------------------------------------------------------ WMMA GEMM ----
// H[N, COLS] = X[N, 64] @ W[64, COLS].  COLS/COLTILES compile-time so store
// offsets fold into immediates. Block = 32 rows x (COLTILES*16) cols; one
// wave per 16x16 tile. W staged in LDS column-major (stride 66, zero-padded
// cols) -> ds_load_2addr_b64 per k-pair, bank-conflict-free. A rows clamped
// (not predicated): invalid rows/cols only pollute masked-out outputs.
template <int COLS, int COLTILES>
__global__ void gemm_wmma_kernel(const float* __restrict__ X,
                                 const float* __restrict__ W,
                                 float* __restrict__ H,
                                 int N) {
    constexpr int KDIM  = 64;
    constexpr int KP    = KDIM + 2;          // LDS column stride (no conflicts)
    constexpr int COLSP = COLTILES * 16;
    __shared__ float sW[COLSP * KP];

    // Stage W (k fastest => conflict-free LDS writes; W is tiny, L2-resident)
    for (int idx = threadIdx.x; idx < COLSP * KDIM; idx += blockDim.x) {
        int n = idx / KDIM;
        int k = idx - n * KDIM;
        sW[n * KP + k] = (n < COLS) ? W[k * COLS + n] : 0.0f;
    }
    __syncthreads();

    const int wave    = threadIdx.x >> 5;
    const int lane    = threadIdx.x & 31;
    const int rowSub  = wave / COLTILES;
    const int colTile = wave - rowSub * COLTILES;
    const int row0 = (blockIdx.x * 2 + rowSub) * 16;
    if (row0 >= N) return;                   // wave-uniform exit (after barrier)

    const int sub = lane >> 4;               // 0: K={0,1}  1: K={2,3} per step
    const int l   = lane & 15;

    int m = row0 + l;                        // A row for this lane (clamped)
    if (m > N - 1) m = N - 1;
    const int n0 = colTile * 16 + l;         // B/D column for this lane

    const float* Xrow = X + (size_t)m * KDIM;
    const float* Wcol = sW + n0 * KP;

    v8f c = {};
#pragma unroll
    for (int k = 0; k < KDIM; k += 4) {
        const int ka = k + 2 * sub;
        v2f a = *(const v2f*)(Xrow + ka);    // global_load_b64 (clause)
        v2f b = *(const v2f*)(Wcol + ka);    // ds_load_2addr_b64
        c = __builtin_amdgcn_wmma_f32_16x16x4_f32(
                false, a, false, b, (short)0, c, false, false);
    }

    if (n0 < COLS) {
        // C/D layout: element c[r] is row (row0 + r + 8*sub), col n0.
        float* Hp = H + (size_t)(row0 + 8 * sub) * COLS + n0;
        if (row0 + 16 <= N) {                // full tile: unguarded store clause
#pragma unroll
            for (int r = 0; r < 8; ++r) Hp[r * COLS] = c[r];
        } else {                             // ragged tail
#pragma unroll
            for (int r = 0; r < 8; ++r)
                if (row0 + 8 * sub + r < N) Hp[r * COLS] = c[r];
        }
    }
}

// ----------------------------------------------- self-loop + bias seeding ---
// out[i,c] = h[i,c] * dinv[i]^2 + bias[c]   (seed before edge atomics)
__global__ void self_bias_kernel(const float* __restrict__ h,
                                 const float* __restrict__ dinv,
                                 const float* __restrict__ bias,
                                 float* __restrict__ out,
                                 int N, int cols) {
    int t = blockIdx.x * blockDim.x + threadIdx.x;
    int total = N * cols;
    if (t >= total) return;
    int i  = t / cols;
    int cc = t - i * cols;
    float di = dinv[i];
    out[t] = h[t] * (di * di) + bias[cc];
}

// -------------------------------------------------------- edge scatter -----
// One thread per (edge, 4-column group): float4 gather of h[src], coef scale,
// 4 x global_atomic_add_f32 into out[dst].
__global__ void scatter_kernel(const long long* __restrict__ eidx,
                               const float* __restrict__ dinv,
                               const float* __restrict__ h,
                               float* __restrict__ out,
                               int E, int cols, int groups) {
    long long tid = (long long)blockIdx.x * blockDim.x + threadIdx.x;
    long long total = (long long)E * groups;
    if (tid >= total) return;
    int e = (int)(tid / groups);
    int g = (int)(tid - (long long)e * groups);

    __builtin_prefetch(eidx + e + 2048, 0, 0);        // global_prefetch_b8

    int s = (int)eidx[e];
    int d = (int)eidx[(size_t)E + e];
    float coef = dinv[s] * dinv[d];

    const float4* hp =
        reinterpret_cast<const float4*>(h + (size_t)s * cols + (size_t)g * 4);
    float4 v = *hp;                                   // 16B aligned (cols 64/40)
    float* op = out + (size_t)d * cols + (size_t)g * 4;
    atomicAdd(op + 0, v.x * coef);
    atomicAdd(op + 1, v.y * coef);
    atomicAdd(op + 2, v.z * coef);
    atomicAdd(op + 3, v.w * coef);
}

// ------------------------------------------------------------------ relu ----
__global__ void relu_kernel(float* __restrict__ x, int n) {
    int i = blockIdx.x * blockDim.x + threadIdx.x;
    if (i < n) x[i] = fmaxf(x[i], 0.0f);
}

// ----------------------------------------------------------- log_softmax ----
// One thread per row; 3 passes over cols (L2-resident).
__global__ void log_softmax_kernel(const float* __restrict__ in,
                                   float* __restrict__ out,
                                   int N, int cols) {
    int i = blockIdx.x * blockDim.x + threadIdx.x;
    if (i >= N) return;
    const float* p = in + (size_t)i * cols;
    float mx = -3.4e38f;
    for (int c = 0; c < cols; ++c) mx = fmaxf(mx, p[c]);
    float s = 0.0f;
    for (int c = 0; c < cols; ++c) s += __expf(p[c] - mx);
    float lse = mx + __logf(s);
    float* q = out + (size_t)i * cols;
    for (int c = 0; c < cols; ++c) q[c] = p[c] - lse;
}

// ================================================================ launch ====
extern "C" void kernel_launch(void* const* d_in, const int* in_sizes, int n_in,
                              void* d_out, int out_size, void* d_ws, size_t ws_size,
                              hipStream_t stream) {
    const float*     x    = (const float*)d_in[0];
    const long long* eidx = (const long long*)d_in[1];   // int64 [2, E]
    const float*     W1   = (const float*)d_in[2];
    const float*     b1   = (const float*)d_in[3];
    const float*     W2   = (const float*)d_in[4];
    const float*     b2   = (const float*)d_in[5];
    float*           out  = (float*)d_out;

    const int Hdim = in_sizes[3];                 // 64
    const int F    = in_sizes[2] / Hdim;          // 64
    const int C    = in_sizes[5];                 // 40
    const int N    = in_sizes[0] / F;             // 100000
    const int E    = in_sizes[1] / 2;             // 1600000

    // workspace layout: dinv | bufA (N x 64) | bufB (N x 64)
    size_t dinvBytes = (((size_t)N * 4) + 511) & ~(size_t)511;
    size_t bufBytes  = (((size_t)N * Hdim * 4) + 511) & ~(size_t)511;
    float* dinv = (float*)d_ws;
    float* bufA = (float*)((char*)d_ws + dinvBytes);
    float* bufB = (float*)((char*)d_ws + dinvBytes + bufBytes);

    const int TPB = 256;

    // --- degree / dinv ---
    deg_init_kernel<<<(N + TPB - 1) / TPB, TPB, 0, stream>>>(dinv, N);
    deg_accum_kernel<<<(E + TPB - 1) / TPB, TPB, 0, stream>>>(eidx, dinv, E);
    dinv_kernel<<<(N + TPB - 1) / TPB, TPB, 0, stream>>>(dinv, N);

    // --- layer 1: h1 = x @ W1 (WMMA, 64 cols -> 4 tiles, 256 thr/blk) ---
    {
        int blocks = (N + 31) / 32;               // 32 rows per block
        gemm_wmma_kernel<64, 4><<<blocks, 256, 0, stream>>>(x, W1, bufA, N);
    }
    // agg1 seeded with self-loop + bias, then edge scatter, then relu
    self_bias_kernel<<<((N * Hdim) + TPB - 1) / TPB, TPB, 0, stream>>>(
        bufA, dinv, b1, bufB, N, Hdim);
    {
        int groups = Hdim / 4;                    // 16
        long long total = (long long)E * groups;
        int blocks = (int)((total + TPB - 1) / TPB);
        scatter_kernel<<<blocks, TPB, 0, stream>>>(eidx, dinv, bufA, bufB,
                                                   E, Hdim, groups);
    }
    relu_kernel<<<((N * Hdim) + TPB - 1) / TPB, TPB, 0, stream>>>(bufB, N * Hdim);

    // --- layer 2: h2 = relu(agg1) @ W2 (WMMA, 40 cols -> 3 masked tiles) ---
    {
        int blocks = (N + 31) / 32;
        gemm_wmma_kernel<40, 3><<<blocks, 192, 0, stream>>>(bufB, W2, bufA, N);
    }
    self_bias_kernel<<<((N * C) + TPB - 1) / TPB, TPB, 0, stream>>>(
        bufA, dinv, b2, bufB, N, C);
    {
        int groups = C / 4;                       // 10
        long long total = (long long)E * groups;
        int blocks = (int)((total + TPB - 1) / TPB);
        scatter_kernel<<<blocks, TPB, 0, stream>>>(eidx, dinv, bufA, bufB,
                                                   E, C, groups);
    }
    relu_kernel<<<((N * C) + TPB - 1) / TPB, TPB, 0, stream>>>(bufB, N * C);

    // --- log_softmax over classes ---
    log_softmax_kernel<<<(N + TPB - 1) / TPB, TPB, 0, stream>>>(bufB, out, N, C);
}